// DivEncLayer_4080218931304
// MI455X (gfx1250) — compile-verified
//
#include <hip/hip_runtime.h>

typedef __attribute__((ext_vector_type(2))) float v2f;
typedef __attribute__((ext_vector_type(4))) float v4f;
typedef __attribute__((ext_vector_type(8))) float v8f;

#define QN 128   // slices
#define SN 8     // slice length (K of layer 1)
#define UN 32    // hidden units (N of layer 1)
#define BT 32    // batch rows staged per block (2 x 16-row WMMA tiles)
#define XS 1028  // padded LDS row stride (dwords) for x tile: 1028%64==4 -> conflict-free A reads
#define OS 132   // padded LDS row stride (dwords) for output staging

__launch_bounds__(256)
__global__ void divenc_mlp_kernel(const float* __restrict__ x,
                                  const float* __restrict__ W1,
                                  const float* __restrict__ b1,
                                  const float* __restrict__ W2,
                                  const float* __restrict__ b2,
                                  float* __restrict__ out)
{
    __shared__ float lds_x[BT * XS];   // 32 x 1024 f32 x-tile (padded)
    __shared__ float lds_o[BT * OS];   // 32 x 128 f32 output staging (padded)

    const int t  = threadIdx.x;
    const int b0 = blockIdx.x * BT;

    // ---- Stage x tile: 32 rows x 1024 floats, coalesced 128-bit non-temporal loads ----
    {
        const v4f* __restrict__ xg =
            reinterpret_cast<const v4f*>(x + (size_t)b0 * (QN * SN));
        #pragma unroll 4
        for (int j = 0; j < (BT * 256) / 256; ++j) {   // 32 iterations
            int vec = j * 256 + t;
            int row = vec >> 8;          // 256 float4 per row
            int c4  = vec & 255;
            v4f v = __builtin_nontemporal_load(&xg[row * 256 + c4]);
            float* dst = &lds_x[row * XS + c4 * 4];
            dst[0] = v.x; dst[1] = v.y; dst[2] = v.z; dst[3] = v.w;
        }
    }
    __syncthreads();

    const int wave = t >> 5;
    const int lane = t & 31;
    const int half = lane >> 4;   // A: K-half select; C/D: M-half select
    const int n    = lane & 15;   // A: M row; B/C/D: N column

    const bool hi8 = (n & 8) != 0;
    const bool hi4 = (n & 4) != 0;
    const bool hi2 = (n & 2) != 0;
    const int  rsel = (hi8 ? 4 : 0) + (hi4 ? 2 : 0) + (hi2 ? 1 : 0); // row this lane ends with

    // Each of 8 waves owns 16 consecutive q slices.
    for (int qi = 0; qi < 16; ++qi) {
        const int q = wave * 16 + qi;

        // ---- B operands (W1[q], 8x32): 2 K-steps x 2 N-tiles, 2 VGPRs each.
        // B(4x16) layout: lane -> N=lane%16, VGPR v -> K = v + 2*(lane/16).
        const float* __restrict__ w1q = W1 + q * (SN * UN);
        const int ko = 2 * half;
        v2f bk0n0 = { w1q[(ko + 0) * UN + n],      w1q[(ko + 1) * UN + n]      };
        v2f bk0n1 = { w1q[(ko + 0) * UN + 16 + n], w1q[(ko + 1) * UN + 16 + n] };
        v2f bk1n0 = { w1q[(ko + 4) * UN + n],      w1q[(ko + 5) * UN + n]      };
        v2f bk1n1 = { w1q[(ko + 4) * UN + 16 + n], w1q[(ko + 5) * UN + 16 + n] };

        const float b1lo = b1[q * UN + n];
        const float b1hi = b1[q * UN + 16 + n];
        const float w2lo = W2[q * UN + n];
        const float w2hi = W2[q * UN + 16 + n];
        const float b2q  = b2[q];

        // Two 16-row batch tiles reuse the weight registers.
        #pragma unroll
        for (int bt = 0; bt < BT / 16; ++bt) {
            // ---- A operands from LDS (16x8 f32 -> two 16x4 K-steps).
            // A layout: lane -> M=lane%16, VGPR v -> K = kbase + v + 2*(lane/16).
            const int arow = bt * 16 + n;
            const float2 a0p = *reinterpret_cast<const float2*>(
                &lds_x[arow * XS + q * SN + 0 + 2 * half]);
            const float2 a1p = *reinterpret_cast<const float2*>(
                &lds_x[arow * XS + q * SN + 4 + 2 * half]);
            v2f a0 = { a0p.x, a0p.y };
            v2f a1 = { a1p.x, a1p.y };

            v8f c0 = {};  // N = 0..15
            v8f c1 = {};  // N = 16..31
            c0 = __builtin_amdgcn_wmma_f32_16x16x4_f32(false, a0, false, bk0n0,
                                                       (short)0, c0, false, false);
            c0 = __builtin_amdgcn_wmma_f32_16x16x4_f32(false, a1, false, bk1n0,
                                                       (short)0, c0, false, false);
            c1 = __builtin_amdgcn_wmma_f32_16x16x4_f32(false, a0, false, bk0n1,
                                                       (short)0, c1, false, false);
            c1 = __builtin_amdgcn_wmma_f32_16x16x4_f32(false, a1, false, bk1n1,
                                                       (short)0, c1, false, false);

            // ---- bias + ELU + layer-2 partial dot per row held by this lane.
            float p[8];
            #pragma unroll
            for (int v = 0; v < 8; ++v) {
                float h0 = c0[v] + b1lo;
                float h1 = c1[v] + b1hi;
                h0 = h0 > 0.0f ? h0 : (__expf(h0) - 1.0f);
                h1 = h1 > 0.0f ? h1 : (__expf(h1) - 1.0f);
                p[v] = h0 * w2lo + h1 * w2hi;
            }

            // ---- value-splitting tree reduction over the 16 N-lanes of each
            // half: 8 shuffles total (vs 32 for per-value butterflies), and
            // each lane ends holding a distinct row's full sum.
            #pragma unroll
            for (int i = 0; i < 4; ++i) {           // mask 8: 8 -> 4 values
                float keep = hi8 ? p[i + 4] : p[i];
                float send = hi8 ? p[i] : p[i + 4];
                p[i] = keep + __shfl_xor(send, 8);
            }
            #pragma unroll
            for (int i = 0; i < 2; ++i) {           // mask 4: 4 -> 2 values
                float keep = hi4 ? p[i + 2] : p[i];
                float send = hi4 ? p[i] : p[i + 2];
                p[i] = keep + __shfl_xor(send, 4);
            }
            {                                       // mask 2: 2 -> 1 value
                float keep = hi2 ? p[1] : p[0];
                float send = hi2 ? p[0] : p[1];
                p[0] = keep + __shfl_xor(send, 2);
            }
            p[0] += __shfl_xor(p[0], 1);            // mask 1: complete 16-lane sum

            // Lane (n even) holds row rsel of its M-half; rows 0..7 covered by
            // n = 0,2,...,14. Bank-conflict-free (bank = 4*row + q mod 64).
            if ((n & 1) == 0)
                lds_o[(bt * 16 + half * 8 + rsel) * OS + q] = p[0] + b2q;
        }
    }
    __syncthreads();

    // ---- Coalesced non-temporal writeout: 32 rows x 128 floats ----
    {
        float* __restrict__ og = out + (size_t)b0 * QN;
        #pragma unroll
        for (int j = 0; j < (BT * QN) / 256; ++j) {   // 16 iterations
            int idx = j * 256 + t;
            int row = idx >> 7;    // /128
            int col = idx & 127;
            __builtin_nontemporal_store(lds_o[row * OS + col], &og[row * QN + col]);
        }
    }
}

extern "C" void kernel_launch(void* const* d_in, const int* in_sizes, int n_in,
                              void* d_out, int out_size, void* d_ws, size_t ws_size,
                              hipStream_t stream) {
    (void)n_in; (void)out_size; (void)d_ws; (void)ws_size;
    const float* x  = (const float*)d_in[0];
    const float* W1 = (const float*)d_in[1];
    const float* b1 = (const float*)d_in[2];
    const float* W2 = (const float*)d_in[3];
    const float* b2 = (const float*)d_in[4];
    float* out = (float*)d_out;

    const int B = in_sizes[0] / (QN * SN);   // 65536
    const int grid = B / BT;                 // 2048 blocks, 256 threads (8 wave32)
    divenc_mlp_kernel<<<grid, 256, 0, stream>>>(x, W1, b1, W2, b2, out);
}